// YOLOLoss_16423954940334
// MI455X (gfx1250) — compile-verified
//
#include <hip/hip_runtime.h>
#include <math.h>

// ---------------- problem constants ----------------
#define NC      80
#define NA      3
#define HW      52
#define NT      50
#define BS      32
#define PLANE   (HW * HW)                 // 2704
#define CELLS   (BS * NA * PLANE)         // 259584
#define NBLK    ((CELLS + 255) / 256)     // 1014 (exact: 1014*256 == CELLS)
#define EPSV    1e-12f

// workspace layout (float units)
#define TP_OFF  16                        // per-target prep: [valid,gx,gy,gw,gh,-,-,-] x (32*50)
#define SC_OFF  12832                     // scatter arrays: smask, stx, sty, stw, sth, scls
#define BP_OFF  (SC_OFF + 6 * CELLS)      // per-block partials: NBLK * 8

typedef float v2f __attribute__((ext_vector_type(2)));
typedef float v8f __attribute__((ext_vector_type(8)));

// anchors / stride = {10,13},{16,30},{33,23} / 8
__constant__ float c_aw[3] = {1.25f, 2.0f,  4.125f};
__constant__ float c_ah[3] = {1.625f, 3.75f, 2.875f};

__device__ __forceinline__ float sigm(float v) { return 1.0f / (1.0f + __expf(-v)); }

// ---------------- kernel 0: zero the scatter mask + n_obj counter ----------------
__global__ void k_zero(float* __restrict__ ws) {
    int idx = blockIdx.x * 256 + threadIdx.x;
    if (idx < CELLS) ws[SC_OFF + idx] = 0.0f;     // smask
    if (idx == 0)    ((int*)ws)[0] = 0;           // n_obj
}

// ---------------- kernel 1: target prep + deterministic scatter ----------------
// one thread per batch, sequential over targets => last-writer-wins is deterministic
__global__ void k_prep(const float* __restrict__ tgt, float* __restrict__ ws) {
    int b = threadIdx.x;
    if (b >= BS) return;
    int cnt = 0;
    for (int t = 0; t < NT; ++t) {
        const float* p = tgt + (b * NT + t) * 5;
        float cls = p[0], cx = p[1], cy = p[2], tw = p[3], th = p[4];
        bool valid = (cls + cx + cy + tw + th) > 0.0f;
        float gx = cx * HW, gy = cy * HW, gw = tw * HW, gh = th * HW;
        int gi = min(max((int)gx, 0), HW - 1);
        int gj = min(max((int)gy, 0), HW - 1);

        float* tp = ws + TP_OFF + (b * NT + t) * 8;
        tp[0] = valid ? 1.0f : 0.0f;
        tp[1] = gx; tp[2] = gy; tp[3] = gw; tp[4] = gh;

        // best anchor by wh-IoU (first-max on ties, like jnp.argmax)
        float best = -1.0f; int bn = 0;
        #pragma unroll
        for (int a = 0; a < 3; ++a) {
            float inter = fminf(gw, c_aw[a]) * fminf(gh, c_ah[a]);
            float uni   = gw * gh + c_aw[a] * c_ah[a] - inter;
            float iou   = inter / (uni + 1e-16f);
            if (iou > best) { best = iou; bn = a; }
        }

        if (valid) {
            ++cnt;
            int idx = ((b * NA + bn) * HW + gj) * HW + gi;
            ws[SC_OFF             + idx] = 1.0f;
            ws[SC_OFF + 1 * CELLS + idx] = gx - (float)gi;
            ws[SC_OFF + 2 * CELLS + idx] = gy - (float)gj;
            ws[SC_OFF + 3 * CELLS + idx] = __logf(gw / c_aw[bn] + 1e-16f);
            ws[SC_OFF + 4 * CELLS + idx] = __logf(gh / c_ah[bn] + 1e-16f);
            ws[SC_OFF + 5 * CELLS + idx] = (float)min(max((int)cls, 0), NC - 1);
        }
    }
    atomicAdd((int*)ws, cnt);   // integer atomic: deterministic result
}

// ---------------- kernel 2: per-cell losses + WMMA block reduction ----------------
__global__ void __launch_bounds__(256) k_main(const float* __restrict__ in,
                                              float* __restrict__ ws) {
    __shared__ float red[256 * 8];

    const int idx = blockIdx.x * 256 + threadIdx.x;   // grid exactly covers CELLS
    const int i = idx % HW;
    const int j = (idx / HW) % HW;
    const int a = (idx / PLANE) % NA;
    const int b = idx / (NA * PLANE);

    const float* base = in + ((size_t)(b * NA * (5 + NC) + a * (5 + NC))) * PLANE
                           + j * HW + i;

    float xr = base[0 * PLANE];
    float yr = base[1 * PLANE];
    float wr = base[2 * PLANE];
    float hr = base[3 * PLANE];
    float cr = base[4 * PLANE];

    float x    = sigm(xr);
    float y    = sigm(yr);
    float conf = sigm(cr);

    // detached predicted box (grid units)
    float aw = c_aw[a], ah = c_ah[a];
    float px = x + (float)i;
    float py = y + (float)j;
    float pw = __expf(wr) * aw;
    float ph = __expf(hr) * ah;
    float b2x1 = px - 0.5f * pw, b2x2 = px + 0.5f * pw;
    float b2y1 = py - 0.5f * ph, b2y2 = py + 0.5f * ph;
    float area2 = pw * ph;

    // ignore mask: any valid gt with IoU >= 0.5 against this predicted box
    bool ignore = false;
    const float* tp0 = ws + TP_OFF + b * NT * 8;
    for (int t = 0; t < NT; ++t) {
        const float* tp = tp0 + t * 8;
        float valid = tp[0];
        float gx = tp[1], gy = tp[2], gw = tp[3], gh = tp[4];
        float b1x1 = gx - 0.5f * gw, b1x2 = gx + 0.5f * gw;
        float b1y1 = gy - 0.5f * gh, b1y2 = gy + 0.5f * gh;
        float iw = fmaxf(fminf(b1x2, b2x2) - fmaxf(b1x1, b2x1), 0.0f);
        float ih = fmaxf(fminf(b1y2, b2y2) - fmaxf(b1y1, b2y1), 0.0f);
        float inter = iw * ih;
        float iou = inter / (gw * gh + area2 - inter + 1e-16f);
        ignore = ignore || ((valid > 0.5f) && (iou >= 0.5f));
    }

    float m   = ws[SC_OFF + idx];
    bool  obj = m > 0.5f;
    float noobj = (ignore || obj) ? 0.0f : 1.0f;

    float comp[8];
    if (obj) {
        float dx = x  - ws[SC_OFF + 1 * CELLS + idx];
        float dy = y  - ws[SC_OFF + 2 * CELLS + idx];
        float dw = wr - ws[SC_OFF + 3 * CELLS + idx];
        float dh = hr - ws[SC_OFF + 4 * CELLS + idx];
        comp[0] = dx * dx; comp[1] = dy * dy; comp[2] = dw * dw; comp[3] = dh * dh;
        comp[4] = -__logf(fmaxf(conf, EPSV));
        // class BCE against one-hot target; only masked cells read class channels
        int ct = (int)ws[SC_OFF + 5 * CELLS + idx];
        float s = 0.0f;
        for (int c = 0; c < NC; ++c) {
            float pc = sigm(base[(5 + c) * PLANE]);
            s += (c == ct) ? -__logf(fmaxf(pc, EPSV))
                           : -__logf(fmaxf(1.0f - pc, EPSV));
        }
        comp[6] = s;
    } else {
        comp[0] = comp[1] = comp[2] = comp[3] = comp[4] = comp[6] = 0.0f;
    }
    comp[5] = noobj * -__logf(fmaxf(1.0f - conf, EPSV));
    comp[7] = (x - 0.5f) * (x - 0.5f) + (y - 0.5f) * (y - 0.5f) + wr * wr + hr * hr;

    #pragma unroll
    for (int c = 0; c < 8; ++c) red[threadIdx.x * 8 + c] = comp[c];
    __syncthreads();

    // ---- WMMA reduction: D = ones(16x4) x B(4x16) + C => column sums of B ----
    // chunk element e (0..63) sits at B[e/16][e%16]; lane l supplies e=l and e=l+32.
    // element index = t*8 + c  =>  column = 8*(t&1) + c, so S[c] = col[c] + col[c+8].
    if (threadIdx.x < 32) {            // whole wave 0, EXEC all ones
        int l = threadIdx.x;
        v8f acc = {};
        v2f ones; ones[0] = 1.0f; ones[1] = 1.0f;
        #pragma unroll
        for (int q = 0; q < 32; ++q) {
            v2f bm;
            bm[0] = red[q * 64 + l];
            bm[1] = red[q * 64 + l + 32];
            acc = __builtin_amdgcn_wmma_f32_16x16x4_f32(
                false, ones, false, bm, (short)0, acc, false, false);
        }
        float d0  = acc[0];                    // row 0: lanes 0..15 hold col sums
        float oth = __shfl(d0, l + 8, 32);
        if (l < 8) ws[BP_OFF + blockIdx.x * 8 + l] = d0 + oth;
    }
}

// ---------------- kernel 3: deterministic fixed-order finalize ----------------
__global__ void k_fin(const float* __restrict__ ws, float* __restrict__ out) {
    __shared__ float S[8];
    int t = threadIdx.x;
    if (t < 8) {
        float s = 0.0f;
        for (int bk = 0; bk < NBLK; ++bk) s += ws[BP_OFF + bk * 8 + t];
        S[t] = s;
    }
    __syncthreads();
    if (t == 0) {
        float n = fmaxf((float)((const int*)ws)[0], 1.0f);
        float loss = (S[0] + S[1] + S[2] + S[3] + 0.1f * S[7]) / (2.0f * n)
                   + 5.0f * (S[4] + 0.2f * S[5]) / n
                   + S[6] / n;
        out[0] = loss;
    }
}

// ---------------- launch ----------------
extern "C" void kernel_launch(void* const* d_in, const int* in_sizes, int n_in,
                              void* d_out, int out_size, void* d_ws, size_t ws_size,
                              hipStream_t stream) {
    const float* inp = (const float*)d_in[0];   // [32, 255, 52, 52]
    const float* tgt = (const float*)d_in[1];   // [32, 50, 5]
    float* out = (float*)d_out;
    float* ws  = (float*)d_ws;

    k_zero<<<NBLK, 256, 0, stream>>>(ws);
    k_prep<<<1, 32, 0, stream>>>(tgt, ws);
    k_main<<<NBLK, 256, 0, stream>>>(inp, ws);
    k_fin <<<1, 32, 0, stream>>>(ws, out);
}